// LSTMCellNM_61924838474038
// MI455X (gfx1250) — compile-verified
//
#include <hip/hip_runtime.h>

#define B_ 256
#define U_ 512
#define D_ 512
#define G4 2048 /* 4*U */
#define CH 4    /* u-chunks per batch in pass 1 */

typedef __attribute__((ext_vector_type(2))) float v2f;
typedef __attribute__((ext_vector_type(4))) float v4f;
typedef __attribute__((ext_vector_type(8))) float v8f;

__device__ __forceinline__ float hsig(float z) {
  return fminf(fmaxf(0.2f * z + 0.5f, 0.0f), 1.0f);
}

// ---------------------------------------------------------------------------
// WMMA fp32 GEMM:  C[M,N] = A[M,K] @ Bm[K,N]  (+ bias[n])
// One wave32 computes one 16x16 tile via V_WMMA_F32_16X16X4_F32, K step 4.
// A 16x4 layout (ISA 7.12.2): lane&15 = M row; lane>>4 selects K pair {0,1}/{2,3}.
// B 4x16: lane&15 = N col; same K-pair split. C/D: v8f, M = vgpr + 8*(lane>=16).
// ---------------------------------------------------------------------------
__global__ void wmma_gemm_f32(const float* __restrict__ A,
                              const float* __restrict__ Bm,
                              const float* __restrict__ bias,
                              float* __restrict__ C,
                              int M, int N, int K, int addBias) {
  const int lane = threadIdx.x & 31;
  const int wave = threadIdx.x >> 5;
  const int tn = blockIdx.x * 4 + wave;  // N tile index (uniform per wave)
  const int tm = blockIdx.y;             // M tile index
  if (tn * 16 >= N || tm * 16 >= M) return;  // wave-uniform: EXEC stays all-1s

  const int r  = lane & 15;         // A row / B col within tile
  const int kh = (lane >> 4) * 2;   // K sub-offset: 0 or 2

  const float* Arow = A + (size_t)(tm * 16 + r) * K;
  const float* Bcol = Bm + (size_t)(tn * 16 + r);

  v8f acc = {};
  for (int k = 0; k < K; k += 4) {
    v2f a, b;
    a.x = Arow[k + kh];
    a.y = Arow[k + kh + 1];
    b.x = Bcol[(size_t)(k + kh) * N];
    b.y = Bcol[(size_t)(k + kh + 1) * N];
    acc = __builtin_amdgcn_wmma_f32_16x16x4_f32(false, a, false, b,
                                                (short)0, acc, false, false);
  }

  const int col = tn * 16 + r;
  const float bb = addBias ? bias[col] : 0.0f;
  const int rowBase = tm * 16 + (lane >> 4) * 8;
  for (int i = 0; i < 8; ++i)
    C[(size_t)(rowBase + i) * N + col] = acc[i] + bb;
}

// ---------------------------------------------------------------------------
// Pass 1: part[b,ch,v] = sum_{u in chunk ch} h_tm1[b,u] * hebb[b,u,v]
// grid (B, CH); 256 threads: two u-rows in flight per iteration (half = t>>7),
// float4 along v. 1024 blocks / 8192 waves keep enough loads outstanding to
// saturate HBM. Half-block accumulators are combined via LDS at the end.
// RT temporal hint on hebb so the tail stays resident in L2 for pass 2.
// ---------------------------------------------------------------------------
__global__ void hebb_dot_kernel(const float* __restrict__ h_tm1,
                                const float* __restrict__ hebb,
                                float* __restrict__ part) {
  __shared__ v4f red[256];
  const int b = blockIdx.x;
  const int ch = blockIdx.y;
  const int t = threadIdx.x;
  const int half = t >> 7;     // 0 or 1: which u-row of the pair
  const int tc = t & 127;      // float4 column within the 512-wide v row
  const int u0 = ch * (U_ / CH);

  const v4f* hb = (const v4f*)(hebb + (size_t)b * U_ * U_);
  const float* hrow = h_tm1 + (size_t)b * U_;

  v4f acc = {};
#pragma unroll 4
  for (int uu = 0; uu < U_ / CH; uu += 2) {
    const int u = u0 + uu + half;
    const int up = (uu + 16 < U_ / CH) ? (u + 16) : u;
    __builtin_prefetch(&hb[(size_t)up * (U_ / 4) + tc], 0, 1);
    const float hu = hrow[u];
    const v4f hv = hb[(size_t)u * (U_ / 4) + tc];
    acc += hu * hv;
  }
  red[t] = acc;
  __syncthreads();
  if (t < 128) {
    const v4f tot = red[t] + red[t + 128];
    ((v4f*)(part + ((size_t)b * CH + ch) * U_))[t] = tot;
  }
}

// ---------------------------------------------------------------------------
// Gates / cell / eta. One block per batch b; 256 threads x 2 v-columns.
// Sums the CH hebb-dot partials, computes i,f,o (with recurrent term),
// itc = tanh(x_c + h@w + alpha*hebbdot), c, h; block-reduces
// eta_b = tanh(h . h2mod); emits g[b,v] = eta*fanout*itc for pass 2.
// ---------------------------------------------------------------------------
__global__ void cell_kernel(const float* __restrict__ gx,
                            const float* __restrict__ rec,
                            const float* __restrict__ hw,
                            const float* __restrict__ part,
                            const float* __restrict__ c_tm1,
                            const float* __restrict__ alpha,
                            const float* __restrict__ h2mod,
                            const float* __restrict__ fanout,
                            float* __restrict__ h_out,
                            float* __restrict__ c_out,
                            float* __restrict__ g_out) {
  __shared__ float red[256];
  __shared__ float s_eta;
  const int b = blockIdx.x;
  const int t = threadIdx.x;
  float partial = 0.0f;
  float itcv[2];
  for (int j = 0; j < 2; ++j) {
    const int v = t + j * 256;
    const size_t gb = (size_t)b * G4 + v;
    const size_t ub = (size_t)b * U_ + v;
    const size_t pb = (size_t)b * CH * U_ + v;
    float hdot = 0.0f;
    for (int cc = 0; cc < CH; ++cc) hdot += part[pb + (size_t)cc * U_];
    const float ii = hsig(gx[gb]          + rec[gb]);
    const float ff = hsig(gx[gb + U_]     + rec[gb + U_]);
    const float oo = hsig(gx[gb + 3 * U_] + rec[gb + 3 * U_]);
    const float itc = tanhf(gx[gb + 2 * U_] + hw[ub] + alpha[v] * hdot);
    const float c = ff * c_tm1[ub] + ii * itc;
    const float hh = oo * tanhf(c);
    c_out[ub] = c;
    h_out[ub] = hh;
    partial += hh * h2mod[v];
    itcv[j] = itc;
  }
  red[t] = partial;
  __syncthreads();
  for (int s = 128; s > 0; s >>= 1) {
    if (t < s) red[t] += red[t + s];
    __syncthreads();
  }
  if (t == 0) s_eta = tanhf(red[0]);
  __syncthreads();
  const float eta = s_eta;
  for (int j = 0; j < 2; ++j) {
    const int v = t + j * 256;
    g_out[(size_t)b * U_ + v] = eta * fanout[v] * itcv[j];
  }
}

// ---------------------------------------------------------------------------
// Pass 2: hebb_new[b,u,v] = clip(hebb[b,u,v] + h_tm1[b,u] * g[b,v], -2, 2)
// 2 (b,u) rows per 256-thread block, float4 per lane. Block order REVERSED so
// first-scheduled blocks touch the hebb tail still resident in L2 from pass 1.
// hebb load is last-use -> NT; hebb_new is write-once -> NT.
// ---------------------------------------------------------------------------
__global__ void hebb_update_kernel(const float* __restrict__ hebb,
                                   const float* __restrict__ h_tm1,
                                   const float* __restrict__ g,
                                   float* __restrict__ hebb_new) {
  const int nPairs = (B_ * U_) / 2;
  const int pair = nPairs - 1 - (int)blockIdx.x;     // reversed traversal
  const int row = pair * 2 + (threadIdx.x >> 7);     // global b*U + u
  const int t = threadIdx.x & 127;
  const int b = row >> 9;
  const float hu = h_tm1[row];
  const v4f gv = ((const v4f*)(g + (size_t)b * U_))[t];
  const v4f* src = (const v4f*)(hebb + (size_t)row * U_);
  v4f* dst = (v4f*)(hebb_new + (size_t)row * U_);
  v4f x = __builtin_nontemporal_load(&src[t]);
  v4f y = x + hu * gv;
  y.x = fminf(fmaxf(y.x, -2.0f), 2.0f);
  y.y = fminf(fmaxf(y.y, -2.0f), 2.0f);
  y.z = fminf(fmaxf(y.z, -2.0f), 2.0f);
  y.w = fminf(fmaxf(y.w, -2.0f), 2.0f);
  __builtin_nontemporal_store(y, &dst[t]);
}

// ---------------------------------------------------------------------------
extern "C" void kernel_launch(void* const* d_in, const int* in_sizes, int n_in,
                              void* d_out, int out_size, void* d_ws, size_t ws_size,
                              hipStream_t stream) {
  (void)in_sizes; (void)n_in; (void)out_size; (void)ws_size;
  const float* x      = (const float*)d_in[0];
  const float* h_tm1  = (const float*)d_in[1];
  const float* c_tm1  = (const float*)d_in[2];
  const float* hebb   = (const float*)d_in[3];
  const float* kern   = (const float*)d_in[4];
  const float* rec_k  = (const float*)d_in[5];
  const float* bias   = (const float*)d_in[6];
  const float* w      = (const float*)d_in[7];
  const float* alpha  = (const float*)d_in[8];
  const float* h2mod  = (const float*)d_in[9];
  const float* fanout = (const float*)d_in[10];

  float* h_out = (float*)d_out;
  float* c_out = h_out + (size_t)B_ * U_;
  float* hebb_out = c_out + (size_t)B_ * U_;

  float* ws   = (float*)d_ws;
  float* gx   = ws;                        // B x 4U : x@kernel + bias
  float* rec  = gx   + (size_t)B_ * G4;    // B x 4U : h@rec_kernel
  float* hw   = rec  + (size_t)B_ * G4;    // B x U  : h@w
  float* part = hw   + (size_t)B_ * U_;    // B x CH x U : hebb-dot partials
  float* g    = part + (size_t)B_ * CH * U_; // B x U : eta*fanout*itc

  // Three WMMA GEMMs (independent; tiny vs. the hebb streaming).
  wmma_gemm_f32<<<dim3(G4 / 64, B_ / 16), dim3(128), 0, stream>>>(
      x, kern, bias, gx, B_, G4, D_, 1);
  wmma_gemm_f32<<<dim3(G4 / 64, B_ / 16), dim3(128), 0, stream>>>(
      h_tm1, rec_k, nullptr, rec, B_, G4, U_, 0);
  wmma_gemm_f32<<<dim3(U_ / 64, B_ / 16), dim3(128), 0, stream>>>(
      h_tm1, w, nullptr, hw, B_, U_, U_, 0);

  // Pass 1 over hebb (256 MB read), 1024 blocks for deep MLP of loads.
  hebb_dot_kernel<<<dim3(B_, CH), dim3(256), 0, stream>>>(h_tm1, hebb, part);

  // Small fused gate/cell/eta pass.
  cell_kernel<<<dim3(B_), dim3(256), 0, stream>>>(
      gx, rec, hw, part, c_tm1, alpha, h2mod, fanout, h_out, c_out, g);

  // Pass 2 over hebb (256 MB read + 256 MB write), reversed for L2 tail reuse.
  hebb_update_kernel<<<dim3((B_ * U_) / 2), dim3(256), 0, stream>>>(
      hebb, h_tm1, g, hebb_out);
}